// TopKRouter_11544872091888
// MI455X (gfx1250) — compile-verified
//
#include <hip/hip_runtime.h>
#include <hip/hip_bf16.h>

// Problem constants (match reference)
#define BT   16384          // B*S tokens
#define HD   2048           // hidden dim
#define NE   64             // experts
#define TOPK 2

// d_out layout (floats), in reference return order
#define W_OFF  0                       // expert_weights  [T,2]
#define I_OFF  (BT * TOPK)             // expert_indices  [T,2] (as float)
#define L_OFF  (2 * BT * TOPK)         // router_logits   [T,64]
#define Z_OFF  (L_OFF + BT * NE)       // z_loss, lb_loss scalars

typedef __attribute__((ext_vector_type(2))) float v2f;
typedef __attribute__((ext_vector_type(8))) float v8f;

// ---------------------------------------------------------------------------
// Kernel 1: router GEMM via V_WMMA_F32_16X16X4_F32
// Each wave: one 16-token M tile x all 64 experts (4 accumulator tiles).
// A fragment (16x4 f32): lane l<16 holds (M=l, K=k..k+1), lane>=16 (M=l-16,
// K=k+2..k+3) -> one b64 load/lane. B[k][n] = W[n][k]; same koff trick gives
// one b64 load/lane per 16x16 expert tile.
// ---------------------------------------------------------------------------
__global__ __launch_bounds__(128)
void router_gemm(const float* __restrict__ X, const float* __restrict__ W,
                 float* __restrict__ logits) {
  const int lane = threadIdx.x & 31;
  const int wave = threadIdx.x >> 5;
  const int m0   = (blockIdx.x * 4 + wave) * 16;
  const int l16  = lane & 15;
  const int koff = (lane >> 4) * 2;

  const float* arow = X + (size_t)(m0 + l16) * HD + koff;
  const float* b0   = W + (size_t)(l16 +  0) * HD + koff;
  const float* b1   = W + (size_t)(l16 + 16) * HD + koff;
  const float* b2   = W + (size_t)(l16 + 32) * HD + koff;
  const float* b3   = W + (size_t)(l16 + 48) * HD + koff;

  v8f c0 = {}, c1 = {}, c2 = {}, c3 = {};

  for (int k = 0; k < HD; k += 16) {
    // prefetch activations one chunk ahead (emits global_prefetch_b8)
    if (k + 512 < HD) __builtin_prefetch(arow + k + 512, 0, 1);
#pragma unroll
    for (int kk = 0; kk < 16; kk += 4) {
      v2f a  = *(const v2f*)(arow + k + kk);
      v2f vb0 = *(const v2f*)(b0 + k + kk);
      v2f vb1 = *(const v2f*)(b1 + k + kk);
      v2f vb2 = *(const v2f*)(b2 + k + kk);
      v2f vb3 = *(const v2f*)(b3 + k + kk);
      c0 = __builtin_amdgcn_wmma_f32_16x16x4_f32(false, a, false, vb0,
                                                 (short)0, c0, false, false);
      c1 = __builtin_amdgcn_wmma_f32_16x16x4_f32(false, a, false, vb1,
                                                 (short)0, c1, false, false);
      c2 = __builtin_amdgcn_wmma_f32_16x16x4_f32(false, a, false, vb2,
                                                 (short)0, c2, false, false);
      c3 = __builtin_amdgcn_wmma_f32_16x16x4_f32(false, a, false, vb3,
                                                 (short)0, c3, false, false);
    }
  }

  // C/D layout: VGPR r -> M = r (lanes 0-15) or r+8 (lanes 16-31), N = l16
  const int rbase = m0 + ((lane >> 4) * 8);
#pragma unroll
  for (int r = 0; r < 8; ++r) {
    float* lp = logits + (size_t)(rbase + r) * NE + l16;
    lp[0]  = c0[r];
    lp[16] = c1[r];
    lp[32] = c2[r];
    lp[48] = c3[r];
  }
}

// ---------------------------------------------------------------------------
// Kernel 2: softmax + top-2 + per-expert statistics (one token per thread)
// ws_acc: [0:64) sum(logit), [64:128) sum(logit^2), [128:192) sum(topk probs)
// ---------------------------------------------------------------------------
__global__ __launch_bounds__(256)
void router_stats(const float* __restrict__ logits, float* __restrict__ out,
                  float* __restrict__ ws_acc) {
  __shared__ float s_sum[NE];
  __shared__ float s_sq[NE];
  __shared__ float s_cnt[NE];

  const int tid  = threadIdx.x;
  const int lane = tid & 31;
  const int t    = blockIdx.x * 256 + tid;

  if (tid < NE) { s_sum[tid] = 0.f; s_sq[tid] = 0.f; s_cnt[tid] = 0.f; }
  __syncthreads();

  float l[NE];
  const float4* lp = (const float4*)(logits + (size_t)t * NE);
#pragma unroll
  for (int i = 0; i < NE / 4; ++i) {
    float4 v = lp[i];
    l[4 * i + 0] = v.x; l[4 * i + 1] = v.y;
    l[4 * i + 2] = v.z; l[4 * i + 3] = v.w;
  }

  // softmax denominator
  float mx = l[0];
#pragma unroll
  for (int e = 1; e < NE; ++e) mx = fmaxf(mx, l[e]);
  float z = 0.f;
#pragma unroll
  for (int e = 0; e < NE; ++e) z += __expf(l[e] - mx);
  const float invz = 1.f / z;

  // top-2 on logits (monotone with probs); ties keep lowest index
  float l1 = l[0], l2 = -3.402823466e38f;
  int   i1 = 0,    i2 = 0;
#pragma unroll
  for (int e = 1; e < NE; ++e) {
    float v = l[e];
    if (v > l1)      { l2 = l1; i2 = i1; l1 = v; i1 = e; }
    else if (v > l2) { l2 = v;  i2 = e; }
  }
  const float p1 = __expf(l1 - mx) * invz;
  const float p2 = __expf(l2 - mx) * invz;
  const float inv12 = 1.f / (p1 + p2);

  out[W_OFF + (size_t)t * 2 + 0] = p1 * inv12;
  out[W_OFF + (size_t)t * 2 + 1] = p2 * inv12;
  out[I_OFF + (size_t)t * 2 + 0] = (float)i1;
  out[I_OFF + (size_t)t * 2 + 1] = (float)i2;

  // per-expert sum / sumsq: wave reduction then LDS atomics (keeps l[] in regs)
#pragma unroll
  for (int e = 0; e < NE; ++e) {
    float v  = l[e];
    float v2 = v * v;
#pragma unroll
    for (int off = 16; off > 0; off >>= 1) {
      v  += __shfl_xor(v,  off, 32);
      v2 += __shfl_xor(v2, off, 32);
    }
    if (lane == 0) {
      atomicAdd(&s_sum[e], v);
      atomicAdd(&s_sq[e],  v2);
    }
  }
  // gathered top-k probs per expert (load balancing)
  atomicAdd(&s_cnt[i1], p1);
  atomicAdd(&s_cnt[i2], p2);
  __syncthreads();

  if (tid < NE) {
    atomicAdd(&ws_acc[tid],          s_sum[tid]);
    atomicAdd(&ws_acc[NE + tid],     s_sq[tid]);
    atomicAdd(&ws_acc[2 * NE + tid], s_cnt[tid]);
  }
}

// ---------------------------------------------------------------------------
// Kernel 3: finalize the two scalar losses (single block, 64 threads)
// ---------------------------------------------------------------------------
__global__ __launch_bounds__(64)
void router_finalize(const float* __restrict__ ws_acc, float* __restrict__ out) {
  __shared__ float red[NE];
  const int e = threadIdx.x;
  const float invT = 1.f / (float)BT;

  const float mean = ws_acc[e] * invT;
  const float var  = ws_acc[NE + e] * invT - mean * mean;

  red[e] = var;
  __syncthreads();
  for (int s = 32; s > 0; s >>= 1) {
    if (e < s) red[e] += red[e + s];
    __syncthreads();
  }
  const float zloss = (red[0] / (float)NE) * 0.001f;
  __syncthreads();

  const float frac = ws_acc[2 * NE + e] * invT;
  red[e] = frac;
  __syncthreads();
  for (int s = 32; s > 0; s >>= 1) {
    if (e < s) red[e] += red[e + s];
    __syncthreads();
  }
  const float meanf = red[0] / (float)NE;
  __syncthreads();

  const float d = frac - meanf;
  red[e] = d * d;
  __syncthreads();
  for (int s = 32; s > 0; s >>= 1) {
    if (e < s) red[e] += red[e + s];
    __syncthreads();
  }
  if (e == 0) {
    const float stdv = sqrtf(red[0] / (float)(NE - 1));   // ddof=1
    const float cv   = stdv / (meanf + 1e-8f);
    out[Z_OFF + 0] = zloss;
    out[Z_OFF + 1] = cv * 0.01f;
  }
}

// ---------------------------------------------------------------------------
extern "C" void kernel_launch(void* const* d_in, const int* in_sizes, int n_in,
                              void* d_out, int out_size, void* d_ws, size_t ws_size,
                              hipStream_t stream) {
  const float* X = (const float*)d_in[0];   // hidden_states [4,4096,2048] f32
  const float* W = (const float*)d_in[1];   // router weight [64,2048] f32
  float* out = (float*)d_out;
  float* acc = (float*)d_ws;                // 192-float accumulator

  hipMemsetAsync(acc, 0, 3 * NE * sizeof(float), stream);

  router_gemm<<<BT / 64, 128, 0, stream>>>(X, W, out + L_OFF);
  router_stats<<<BT / 256, 256, 0, stream>>>(out + L_OFF, out, acc);
  router_finalize<<<1, NE, 0, stream>>>(acc, out);
}